// CSR_9569187136084
// MI455X (gfx1250) — compile-verified
//
#include <hip/hip_runtime.h>

typedef __attribute__((ext_vector_type(2))) float v2f;
typedef __attribute__((ext_vector_type(8))) float v8f;

#define DIN 256
#define DD 64
#define BB 32
#define LLEN 200
#define PLN 8
#define SLN 8
#define ROWS_SEQ (BB * LLEN)                 /* 6400 */
#define ROWS_P   (BB * PLN)                  /* 256  */
#define ROWS_TOT (ROWS_SEQ + 2 * ROWS_P)     /* 6912 */
#define EPSF  1e-8f
#define LNEPS 1e-5f

// ---------------------------------------------------------------------------
// Phase 1: fused Q/K/V projection GEMM using V_WMMA_F32_16X16X4_F32.
// Each wave computes one 16-row slab of all three [rows x 64] projections.
// A fragment (16x4 f32): lanes 0-15 hold (row=lane, K=k..k+1), lanes 16-31
// hold (row=lane-16, K=k+2..k+3) -> contiguous float2 per lane.
// B fragment (4x16 f32): symmetric layout from W[n][k..] (W is [64,256]).
// ---------------------------------------------------------------------------
__global__ __launch_bounds__(256) void proj_wmma_kernel(
    const float* __restrict__ seqs, const float* __restrict__ P,
    const float* __restrict__ S,
    const float* __restrict__ Wq, const float* __restrict__ bq,
    const float* __restrict__ Wk, const float* __restrict__ bk,
    const float* __restrict__ Wv, const float* __restrict__ bv,
    float* __restrict__ Qo, float* __restrict__ Zo, float* __restrict__ Vo)
{
  const int lane = threadIdx.x & 31;
  const int wave = threadIdx.x >> 5;
  const int slab = blockIdx.x * 8 + wave;
  const int r0   = slab * 16;

  const float* src;
  if (r0 < ROWS_SEQ)                src = seqs + (size_t)r0 * DIN;
  else if (r0 < ROWS_SEQ + ROWS_P)  src = P + (size_t)(r0 - ROWS_SEQ) * DIN;
  else                              src = S + (size_t)(r0 - ROWS_SEQ - ROWS_P) * DIN;

  const int arow = lane & 15;          // row (A) / col-within-tile (B,C)
  const int koff = (lane >> 4) << 1;   // 0 for lanes 0-15, 2 for lanes 16-31

  const float* Ws[3] = {Wq, Wk, Wv};
  const float* bs[3] = {bq, bk, bv};
  float*       Os[3] = {Qo, Zo, Vo};

  v8f acc[3][4] = {};
  const float* aptr = src + (size_t)arow * DIN + koff;

  for (int kk = 0; kk < DIN; kk += 4) {
    v2f a = *(const v2f*)(aptr + kk);
#pragma unroll
    for (int m = 0; m < 3; ++m) {
      const float* wb = Ws[m] + (size_t)arow * DIN + koff + kk;
#pragma unroll
      for (int n = 0; n < 4; ++n) {
        v2f bf = *(const v2f*)(wb + (size_t)n * 16 * DIN);
        acc[m][n] = __builtin_amdgcn_wmma_f32_16x16x4_f32(
            false, a, false, bf, (short)0, acc[m][n], false, false);
      }
    }
  }

  // C/D layout: VGPR r holds rows r (lanes 0-15) and r+8 (lanes 16-31).
  const int rbase = r0 + ((lane >> 4) << 3);
#pragma unroll
  for (int m = 0; m < 3; ++m) {
#pragma unroll
    for (int n = 0; n < 4; ++n) {
      const int col = n * 16 + arow;
      const float bias = bs[m][col];
#pragma unroll
      for (int r = 0; r < 8; ++r) {
        float v = acc[m][n][r] + bias;
        if (m < 2) v = (v > 0.f) ? v : (expf(v) - 1.f);   // ELU for q, z
        Os[m][(size_t)(rbase + r) * DD + col] = v;
      }
    }
  }
}

// ---------------------------------------------------------------------------
// Reductions (wave32)
// ---------------------------------------------------------------------------
__device__ __forceinline__ float waveRed(float x) {
#pragma unroll
  for (int o = 16; o > 0; o >>= 1) x += __shfl_xor(x, o);
  return x;
}

__device__ __forceinline__ void blockReduce2(float& a, float& b, float* red) {
  const int t = threadIdx.x;
#pragma unroll
  for (int o = 16; o > 0; o >>= 1) { a += __shfl_xor(a, o); b += __shfl_xor(b, o); }
  const int w = t >> 5;
  if ((t & 31) == 0) { red[w] = a; red[8 + w] = b; }
  __syncthreads();
  if (t == 0) {
    float sa = 0.f, sb = 0.f;
#pragma unroll
    for (int i = 0; i < 8; ++i) { sa += red[i]; sb += red[8 + i]; }
    red[16] = sa; red[17] = sb;
  }
  __syncthreads();
  a = red[16]; b = red[17];
}

// ---------------------------------------------------------------------------
// Phase 2: one workgroup per batch element; sequential scan over L with the
// 64x64 running state held in registers (16 elems/thread, 256 threads).
// Thread t owns s elements i = t + 256*j  ->  p = (t>>6)+4*j, q = t&63.
// ---------------------------------------------------------------------------
__global__ __launch_bounds__(256) void scan_kernel(
    const int* __restrict__ user_id,
    const float* __restrict__ prev_z, const float* __restrict__ prev_s,
    const float* __restrict__ Qw, const float* __restrict__ Zw,
    const float* __restrict__ Vw,
    float* __restrict__ out, float* __restrict__ accum)
{
  __shared__ float tmpZ[PLN * DD], tmpV[PLN * DD];
  __shared__ float pzS[DD], PzLast[DD], cum_z[DD];
  __shared__ float zl[DD], vlS[DD], qlS[DD], nqS[DD], naS[DD];
  __shared__ float naPart[256];
  __shared__ float red[18];
  __shared__ float scal[4];   // [0]=||z_final|| per step  [1]=mask  [2]=||zu||

  const int t     = threadIdx.x;
  const int b     = blockIdx.x;
  const int lane  = t & 31;
  const int wid   = t >> 5;
  const int qcol  = t & 63;
  const int prow0 = t >> 6;

  const int uid = user_id[b];

  if (t < DD) { pzS[t] = prev_z[(size_t)uid * DD + t]; cum_z[t] = 0.f; }
  const size_t rowP0 = (size_t)(ROWS_SEQ + b * PLN) * DD;
  for (int i = t; i < PLN * DD; i += 256) { tmpZ[i] = Zw[rowP0 + i]; tmpV[i] = Vw[rowP0 + i]; }
  __syncthreads();

  if (wid == 0) {                                  // history mask
    float s = waveRed(pzS[lane] + pzS[lane + 32]);
    if (lane == 0) scal[1] = (s != 0.f) ? 1.f : 0.f;
  }
  if (t < DD) {                                    // Pz raw (sum over PL)
    float s = 0.f;
    for (int l2 = 0; l2 < PLN; ++l2) s += tmpZ[l2 * DD + t];
    PzLast[t] = s;
  }
  __syncthreads();
  if (wid == 0) {                                  // LN(Pz) over D
    float v0 = PzLast[lane], v1 = PzLast[lane + 32];
    float s  = waveRed(v0 + v1);
    float ss = waveRed(v0 * v0 + v1 * v1);
    float m  = s * (1.f / 64.f);
    float w  = rsqrtf(ss * (1.f / 64.f) - m * m + LNEPS);
    PzLast[lane] = (v0 - m) * w;
    PzLast[lane + 32] = (v1 - m) * w;
  }

  float psLast[16];                                // LN(Ps) over DxD -> regs
  {
    float su = 0.f, ssq = 0.f;
#pragma unroll
    for (int j = 0; j < 16; ++j) {
      const int p = prow0 + 4 * j;
      float s = 0.f;
      for (int l2 = 0; l2 < PLN; ++l2) s += tmpZ[l2 * DD + p] * tmpV[l2 * DD + qcol];
      psLast[j] = s; su += s; ssq += s * s;
    }
    blockReduce2(su, ssq, red);
    float m = su * (1.f / 4096.f);
    float w = rsqrtf(ssq * (1.f / 4096.f) - m * m + LNEPS);
#pragma unroll
    for (int j = 0; j < 16; ++j) psLast[j] = (psLast[j] - m) * w;
  }
  __syncthreads();

  // --- stability prompt S -> user recurrent state ---
  const size_t rowS0 = (size_t)(ROWS_SEQ + ROWS_P + b * SLN) * DD;
  for (int i = t; i < SLN * DD; i += 256) { tmpZ[i] = Zw[rowS0 + i]; tmpV[i] = Vw[rowS0 + i]; }
  __syncthreads();
  if (t < DD) {
    float s = 0.f;
    for (int l2 = 0; l2 < SLN; ++l2) s += tmpZ[l2 * DD + t];
    naS[t] = s;
  }
  __syncthreads();
  if (wid == 0) {                                  // zu = pz + LN(Sz); ||zu||
    float v0 = naS[lane], v1 = naS[lane + 32];
    float s  = waveRed(v0 + v1);
    float ss = waveRed(v0 * v0 + v1 * v1);
    float m  = s * (1.f / 64.f);
    float w  = rsqrtf(ss * (1.f / 64.f) - m * m + LNEPS);
    float zu0 = pzS[lane] + (v0 - m) * w;
    float zu1 = pzS[lane + 32] + (v1 - m) * w;
    float nn = waveRed(zu0 * zu0 + zu1 * zu1);
    if (lane == 0) scal[2] = sqrtf(nn);
  }
  __syncthreads();

  float prevNAMr[16];                              // mask * su / (||zu||+eps)
  {
    float su = 0.f, ssq = 0.f;
    float ssj[16];
#pragma unroll
    for (int j = 0; j < 16; ++j) {
      const int p = prow0 + 4 * j;
      float s = 0.f;
      for (int l2 = 0; l2 < SLN; ++l2) s += tmpZ[l2 * DD + p] * tmpV[l2 * DD + qcol];
      ssj[j] = s; su += s; ssq += s * s;
    }
    blockReduce2(su, ssq, red);
    float m = su * (1.f / 4096.f);
    float w = rsqrtf(ssq * (1.f / 4096.f) - m * m + LNEPS);
    const float invz = scal[1] / (scal[2] + EPSF);
#pragma unroll
    for (int j = 0; j < 16; ++j)
      prevNAMr[j] = (prev_s[(size_t)uid * 4096 + t + 256 * j] + (ssj[j] - m) * w) * invz;
  }

  float cs[16];
#pragma unroll
  for (int j = 0; j < 16; ++j) cs[j] = 0.f;
  float stdSum = 0.f, stdSS = 0.f;

  // ----------------------- sequential scan over L --------------------------
  for (int l = 0; l < LLEN; ++l) {
    __syncthreads();
    const size_t base = (size_t)(b * LLEN + l) * DD;
    if (t < 64)       zl[t]        = Zw[base + t];
    else if (t < 128) vlS[t - 64]  = Vw[base + t - 64];
    else if (t < 192) qlS[t - 128] = Qw[base + t - 128];
    __syncthreads();

    if (wid == 0) {                     // z path: cumsum, LN, +Pz, norm, std
      float c0 = cum_z[lane] + zl[lane];           cum_z[lane] = c0;
      float c1 = cum_z[lane + 32] + zl[lane + 32]; cum_z[lane + 32] = c1;
      float s  = waveRed(c0 + c1);
      float ss = waveRed(c0 * c0 + c1 * c1);
      float m  = s * (1.f / 64.f);
      float w  = rsqrtf(ss * (1.f / 64.f) - m * m + LNEPS);
      float z0 = (c0 - m) * w + PzLast[lane];
      float z1 = (c1 - m) * w + PzLast[lane + 32];
      float nn = waveRed(z0 * z0 + z1 * z1);
      if (lane == 0) scal[0] = sqrtf(nn);
      float r0v = qlS[lane]      / fmaxf(z0, 1e-6f);
      float r1v = qlS[lane + 32] / fmaxf(z1, 1e-6f);
      stdSum += r0v + r1v;
      stdSS  += r0v * r0v + r1v * r1v;
    } else if (wid == 1) {              // nq = q / (||q||+eps)
      float q0 = qlS[lane], q1 = qlS[lane + 32];
      float nn = waveRed(q0 * q0 + q1 * q1);
      float inv = 1.f / (sqrtf(nn) + EPSF);
      nqS[lane] = q0 * inv; nqS[lane + 32] = q1 * inv;
    }
    __syncthreads();

    // s path: cum_s += z (x) v ; LN ; +Ps ; /||z|| folded into second LN
    const float vq = vlS[qcol];
    float s1 = 0.f, ss1 = 0.f;
#pragma unroll
    for (int j = 0; j < 16; ++j) {
      float nv = cs[j] + zl[prow0 + 4 * j] * vq;
      cs[j] = nv; s1 += nv; ss1 += nv * nv;
    }
    blockReduce2(s1, ss1, red);
    float m1 = s1 * (1.f / 4096.f);
    float w1 = rsqrtf(ss1 * (1.f / 4096.f) - m1 * m1 + LNEPS);

    float s2v[16];
    float s2 = 0.f, ss2 = 0.f;
#pragma unroll
    for (int j = 0; j < 16; ++j) {
      float v = (cs[j] - m1) * w1 + psLast[j];
      s2v[j] = v; s2 += v; ss2 += v * v;
    }
    blockReduce2(s2, ss2, red);
    float m2   = s2 * (1.f / 4096.f);
    float var2 = ss2 * (1.f / 4096.f) - m2 * m2;
    float c    = scal[0] + EPSF;
    float w2   = rsqrtf(var2 / (c * c) + LNEPS) / c;   // folded /||z|| LN

    float pna = 0.f;
#pragma unroll
    for (int j = 0; j < 16; ++j)
      pna += nqS[prow0 + 4 * j] * ((s2v[j] - m2) * w2 + prevNAMr[j]);
    naPart[t] = pna;
    __syncthreads();
    if (t < 64) naS[t] = naPart[t] + naPart[t + 64] + naPart[t + 128] + naPart[t + 192];
    __syncthreads();
    if (wid == 0) {                      // final LN over D, write NA
      float v0 = naS[lane], v1 = naS[lane + 32];
      float s  = waveRed(v0 + v1);
      float ss = waveRed(v0 * v0 + v1 * v1);
      float m  = s * (1.f / 64.f);
      float w  = rsqrtf(ss * (1.f / 64.f) - m * m + LNEPS);
      out[base + lane]      = (v0 - m) * w;
      out[base + lane + 32] = (v1 - m) * w;
    }
  }

  if (wid == 0) {                        // per-block std partials
    stdSum = waveRed(stdSum);
    stdSS  = waveRed(stdSS);
    if (lane == 0) { accum[2 * b] = stdSum; accum[2 * b + 1] = stdSS; }
  }
}

// ---------------------------------------------------------------------------
// Phase 3: unbiased std over all B*L*D elements.
// ---------------------------------------------------------------------------
__global__ void finalize_kernel(const float* __restrict__ accum,
                                float* __restrict__ out)
{
  if (threadIdx.x == 0 && blockIdx.x == 0) {
    float s = 0.f, ss = 0.f;
    for (int i = 0; i < BB; ++i) { s += accum[2 * i]; ss += accum[2 * i + 1]; }
    const float N = (float)(BB * LLEN * DD);
    float var = (ss - s * s / N) / (N - 1.f);
    out[BB * LLEN * DD] = sqrtf(fmaxf(var, 0.f));
  }
}

// ---------------------------------------------------------------------------
extern "C" void kernel_launch(void* const* d_in, const int* in_sizes, int n_in,
                              void* d_out, int out_size, void* d_ws, size_t ws_size,
                              hipStream_t stream)
{
  const int*   user_id = (const int*)d_in[0];
  const float* seqs    = (const float*)d_in[1];
  const float* P       = (const float*)d_in[2];
  const float* S       = (const float*)d_in[3];
  const float* prev_z  = (const float*)d_in[4];
  const float* prev_s  = (const float*)d_in[5];
  const float* Wq      = (const float*)d_in[6];
  const float* bq      = (const float*)d_in[7];
  const float* Wk      = (const float*)d_in[8];
  const float* bk      = (const float*)d_in[9];
  const float* Wv      = (const float*)d_in[10];
  const float* bv      = (const float*)d_in[11];

  float* out = (float*)d_out;
  float* ws  = (float*)d_ws;
  float* Qw    = ws;
  float* Zw    = ws + (size_t)ROWS_TOT * DD;
  float* Vw    = ws + (size_t)2 * ROWS_TOT * DD;
  float* accum = ws + (size_t)3 * ROWS_TOT * DD;

  proj_wmma_kernel<<<ROWS_TOT / 16 / 8, 256, 0, stream>>>(
      seqs, P, S, Wq, bq, Wk, bk, Wv, bv, Qw, Zw, Vw);
  scan_kernel<<<BB, 256, 0, stream>>>(user_id, prev_z, prev_s, Qw, Zw, Vw, out, accum);
  finalize_kernel<<<1, 32, 0, stream>>>(accum, out);
}